// QuantumConv1d_28467043237966
// MI455X (gfx1250) — compile-verified
//
#include <hip/hip_runtime.h>
#include <hip/hip_bf16.h>

typedef __attribute__((ext_vector_type(2))) float v2f;
typedef __attribute__((ext_vector_type(8))) float v8f;

#define BATCH   32
#define CIN     80
#define LEN     3000
#define OUTCH   512
#define KW      3
#define NQ      4
#define QDIM    16
#define TILE_L  256   // positions per block (16 WMMA column-tiles)
#define OTILES  (OUTCH / 16)   // 32
#define NTILES  (TILE_L / 16)  // 16

// ---- 16-amplitude state-vector helpers (all indices constant-folded) ----
template <int BIT>
__device__ __forceinline__ void apply1q(float (&sr)[QDIM], float (&si)[QDIM],
                                        float u00r, float u00i, float u01r, float u01i,
                                        float u10r, float u10i, float u11r, float u11i) {
#pragma unroll
    for (int i0 = 0; i0 < QDIM; ++i0) {
        if ((i0 & BIT) == 0) {
            const int i1 = i0 | BIT;
            const float ar = sr[i0], ai = si[i0];
            const float br = sr[i1], bi = si[i1];
            sr[i0] = u00r * ar - u00i * ai + u01r * br - u01i * bi;
            si[i0] = u00r * ai + u00i * ar + u01r * bi + u01i * br;
            sr[i1] = u10r * ar - u10i * ai + u11r * br - u11i * bi;
            si[i1] = u10r * ai + u10i * ar + u11r * bi + u11i * br;
        }
    }
}

template <int CBIT, int TBIT>
__device__ __forceinline__ void applycnot(float (&sr)[QDIM], float (&si)[QDIM]) {
#pragma unroll
    for (int i = 0; i < QDIM; ++i) {
        if ((i & CBIT) != 0 && (i & TBIT) == 0) {
            const int j = i | TBIT;
            const float tr = sr[i], ti = si[i];
            sr[i] = sr[j]; si[i] = si[j];
            sr[j] = tr;    si[j] = ti;
        }
    }
}

__global__ void __launch_bounds__(256)
qconv1d_kernel(const float* __restrict__ x,      // (B, C, L)
               const float* __restrict__ W_pre,  // (4, 240)
               const float* __restrict__ b_pre,  // (4,)
               const float* __restrict__ W_post, // (512, 4)
               const float* __restrict__ b_post, // (512,)
               const float* __restrict__ qw,     // (4, 3)
               float* __restrict__ out)          // (B, 512, L)
{
    __shared__ float s_expz[TILE_L * NQ];        // 4 KB
    __shared__ float s_wpre[NQ * CIN * KW];      // 3.75 KB

    const int b   = blockIdx.y;
    const int l0  = blockIdx.x * TILE_L;
    const int tid = threadIdx.x;

    // Stage W_pre into LDS (row-major (4, 240))
    for (int i = tid; i < NQ * CIN * KW; i += blockDim.x) s_wpre[i] = W_pre[i];
    __syncthreads();

    // ------------------- Phase 1: per-position pre -> quantum -> <Z> -------------------
    {
        const int l = l0 + tid;
        float e0 = 0.f, e1 = 0.f, e2 = 0.f, e3 = 0.f;
        if (l < LEN) {
            float p0 = b_pre[0], p1 = b_pre[1], p2 = b_pre[2], p3 = b_pre[3];
            const float* xb = x + (size_t)b * CIN * LEN;
            const bool lo = (l - 1 >= 0), hi = (l + 1 < LEN);
#pragma unroll 4
            for (int c = 0; c < CIN; ++c) {
                const float* xc = xb + (size_t)c * LEN;
                const float x0 = lo ? xc[l - 1] : 0.f;
                const float x1 = xc[l];
                const float x2 = hi ? xc[l + 1] : 0.f;
                const float* w = s_wpre + c * KW;
                p0 += x0 * w[0]       + x1 * w[1]       + x2 * w[2];
                p1 += x0 * w[240 + 0] + x1 * w[240 + 1] + x2 * w[240 + 2];
                p2 += x0 * w[480 + 0] + x1 * w[480 + 1] + x2 * w[480 + 2];
                p3 += x0 * w[720 + 0] + x1 * w[720 + 1] + x2 * w[720 + 2];
            }
            const float invn = rsqrtf(p0 * p0 + p1 * p1 + p2 * p2 + p3 * p3);

            float sr[QDIM], si[QDIM];
#pragma unroll
            for (int i = 0; i < QDIM; ++i) { sr[i] = 0.f; si[i] = 0.f; }
            sr[0] = p0 * invn; sr[1] = p1 * invn; sr[2] = p2 * invn; sr[3] = p3 * invn;

            // Rotations: U = Rz(omega)·Ry(theta)·Rz(phi) per reference _rot_mats
#pragma unroll
            for (int q = 0; q < NQ; ++q) {
                const float phi = qw[q * 3 + 0], theta = qw[q * 3 + 1], omega = qw[q * 3 + 2];
                const float ct = __cosf(0.5f * theta), st = __sinf(0.5f * theta);
                const float a  = 0.5f * (phi + omega), d = 0.5f * (phi - omega);
                const float ca = __cosf(a), sa = __sinf(a);
                const float cd = __cosf(d), sd = __sinf(d);
                // U00=e^{-ia}c, U01=-e^{+id}s, U10=e^{-id}s, U11=e^{+ia}c
                const float u00r =  ca * ct, u00i = -sa * ct;
                const float u01r = -cd * st, u01i = -sd * st;
                const float u10r =  cd * st, u10i = -sd * st;
                const float u11r =  ca * ct, u11i =  sa * ct;
                if (q == 0) apply1q<8>(sr, si, u00r, u00i, u01r, u01i, u10r, u10i, u11r, u11i);
                if (q == 1) apply1q<4>(sr, si, u00r, u00i, u01r, u01i, u10r, u10i, u11r, u11i);
                if (q == 2) apply1q<2>(sr, si, u00r, u00i, u01r, u01i, u10r, u10i, u11r, u11i);
                if (q == 3) apply1q<1>(sr, si, u00r, u00i, u01r, u01i, u10r, u10i, u11r, u11i);
            }
            applycnot<8, 4>(sr, si);
            applycnot<4, 2>(sr, si);
            applycnot<2, 1>(sr, si);

            // expz[q] = sum_i |s_i|^2 * (1 - 2*bit_q(i))
#pragma unroll
            for (int i = 0; i < QDIM; ++i) {
                const float p = sr[i] * sr[i] + si[i] * si[i];
                e0 += (i & 8) ? -p : p;
                e1 += (i & 4) ? -p : p;
                e2 += (i & 2) ? -p : p;
                e3 += (i & 1) ? -p : p;
            }
        }
        s_expz[tid * NQ + 0] = e0;
        s_expz[tid * NQ + 1] = e1;
        s_expz[tid * NQ + 2] = e2;
        s_expz[tid * NQ + 3] = e3;
    }
    __syncthreads();

    // ------------------- Phase 2: out = W_post @ expz + b_post via WMMA f32 16x16x4 ----
    // A (16x4): lane m (0..15) VGPRs = {K0,K1}; lane 16+m VGPRs = {K2,K3}
    // B (4x16): lane m = col n; lane half selects K pair {0,1} vs {2,3}
    // C/D (16x16): VGPR r, lanes 0-15 -> M=r, lanes 16-31 -> M=r+8
    const int wave = tid >> 5;
    const int lane = tid & 31;
    const int m    = lane & 15;
    const int kh   = lane >> 4;           // 0 or 1

    for (int ot = wave; ot < OTILES; ot += 8) {
        const int o0 = ot * 16;
        // A fragment from W_post (row-major (512,4))
        v2f afrag;
        {
            const float* wp = W_post + (size_t)(o0 + m) * NQ + kh * 2;
            afrag.x = wp[0];
            afrag.y = wp[1];
        }
        // bias per accumulator row: row(r) = o0 + r + kh*8
        float bias[8];
#pragma unroll
        for (int r = 0; r < 8; ++r) bias[r] = b_post[o0 + r + kh * 8];

        for (int nt = 0; nt < NTILES; ++nt) {
            const int n0 = nt * 16;
            v2f bfrag;
            {
                const float* ez = s_expz + (n0 + m) * NQ + kh * 2;
                bfrag.x = ez[0];
                bfrag.y = ez[1];
            }
            v8f acc;
#pragma unroll
            for (int r = 0; r < 8; ++r) acc[r] = bias[r];

            acc = __builtin_amdgcn_wmma_f32_16x16x4_f32(
                /*neg_a=*/false, afrag, /*neg_b=*/false, bfrag,
                /*c_mod=*/(short)0, acc, /*reuse_a=*/false, /*reuse_b=*/false);

            const int lcol = l0 + n0 + m;      // this lane's output column
            if (lcol < LEN) {
                float* op = out + ((size_t)b * OUTCH + (o0 + kh * 8)) * LEN + lcol;
#pragma unroll
                for (int r = 0; r < 8; ++r) op[(size_t)r * LEN] = acc[r];
            }
        }
    }
}

extern "C" void kernel_launch(void* const* d_in, const int* in_sizes, int n_in,
                              void* d_out, int out_size, void* d_ws, size_t ws_size,
                              hipStream_t stream) {
    const float* x      = (const float*)d_in[0];
    const float* W_pre  = (const float*)d_in[1];
    const float* b_pre  = (const float*)d_in[2];
    const float* W_post = (const float*)d_in[3];
    const float* b_post = (const float*)d_in[4];
    const float* qw     = (const float*)d_in[5];
    float* out          = (float*)d_out;

    dim3 grid((LEN + TILE_L - 1) / TILE_L, BATCH);   // (12, 32)
    qconv1d_kernel<<<grid, 256, 0, stream>>>(x, W_pre, b_pre, W_post, b_post, qw, out);
}